// PositionalEncoding_84284438216797
// MI455X (gfx1250) — compile-verified
//
#include <hip/hip_runtime.h>

// out[b,s,d] = x[b,s,d] + pe[s,d]
//   pe[s,2i]   = sin(s * 10000^(-2i/D))
//   pe[s,2i+1] = cos(s * 10000^(-2i/D))
// B=4, S=4096, D=2048. Pure streaming add: 256 MiB traffic -> HBM-bound
// (~11.5 us at 23.3 TB/s). PE computed once per (s,d) float4, reused over
// all 4 batch slabs. Loads RT (keep 128MB x resident in 192MB L2 across
// replays), stores NT (don't let the write stream evict x).

#define PE_B 4
#define PE_S 4096
#define PE_D 2048
#define PE_D4 (PE_D / 4)          // 512 float4 per row
#define PE_SLAB (PE_S * PE_D)     // elements per batch slab

typedef __attribute__((ext_vector_type(4))) float floatx4;

__global__ __launch_bounds__(256) void
PositionalEncoding_84284438216797_kernel(const float* __restrict__ x,
                                         float* __restrict__ out) {
    // One thread per float4 of one (s,d) row position; covers all 4 batches.
    const int t = blockIdx.x * 256 + threadIdx.x;   // [0, PE_S*PE_D4)
    const int c = t & (PE_D4 - 1);                  // float4 column, 0..511
    const int s = t >> 9;                           // sequence position

    // This float4 covers d = 4c..4c+3 -> pair indices i0 = 2c (sin/cos),
    // i1 = 2c+1 (sin/cos).
    // angle = s * exp2( -(2*i/D) * log2(10000) )
    const float kExp = -(2.0f / (float)PE_D) * 13.28771237954945f; // -2/D*log2(1e4)
    const float fs   = (float)s;
    const float a0   = fs * __builtin_exp2f(kExp * (float)(2 * c));
    const float a1   = fs * __builtin_exp2f(kExp * (float)(2 * c + 1));

    // Range-reduce once in revolution domain, shared by v_sin/v_cos.
    const float INV2PI = 0.15915494309189535f;
    float r0 = a0 * INV2PI; r0 -= __builtin_floorf(r0);
    float r1 = a1 * INV2PI; r1 -= __builtin_floorf(r1);

    floatx4 pe;
    pe.x = __builtin_amdgcn_sinf(r0);   // v_sin_f32
    pe.y = __builtin_amdgcn_cosf(r0);   // v_cos_f32
    pe.z = __builtin_amdgcn_sinf(r1);
    pe.w = __builtin_amdgcn_cosf(r1);

    const int off = s * PE_D + 4 * c;   // element offset inside a slab

#pragma unroll
    for (int b = 0; b < PE_B; ++b) {
        const int o = off + b * PE_SLAB;
        // Regular (RT) b128 load: x should stay L2-resident across replays.
        floatx4 v = *(const floatx4*)(x + o);
        v = v + pe;
        // NT b128 store: write-once stream, keep it out of x's L2 footprint.
        __builtin_nontemporal_store(v, (floatx4*)(out + o));
    }
}

extern "C" void kernel_launch(void* const* d_in, const int* in_sizes, int n_in,
                              void* d_out, int out_size, void* d_ws, size_t ws_size,
                              hipStream_t stream) {
    (void)in_sizes; (void)n_in; (void)d_ws; (void)ws_size; (void)out_size;
    const float* x = (const float*)d_in[0];
    float* out = (float*)d_out;

    const int threads = PE_S * PE_D4;           // 2,097,152
    const int block = 256;                      // 8 wave32s
    const int grid = threads / block;           // 8192, exact
    PositionalEncoding_84284438216797_kernel<<<grid, block, 0, stream>>>(x, out);
}